// ProposalLayer_61701500174896
// MI455X (gfx1250) — compile-verified
//
#include <hip/hip_runtime.h>
#include <hip/hip_bf16.h>
#include <stdint.h>

#define B_IMG   16
#define FH      192
#define FW      192
#define A_ANCH  9
#define N_ANCH  (FH * FW * A_ANCH)   // 331776
#define P_PRE   6000
#define Q_POST  1000
#define IOU_THR 0.7f
#define NEG_V   (-1e9f)

// radix split of the 32-bit monotone key: 11 + 12 + 9 bits
#define L1_BINS 2048
#define L2_BINS 4096
#define L3_BINS 512

// ---------------- workspace layout (bytes) ----------------
static constexpr size_t WS_H1   = 0;                                      // B*2048 u32
static constexpr size_t WS_H2   = WS_H1 + (size_t)B_IMG * L1_BINS * 4;    // B*4096 u32
static constexpr size_t WS_H3   = WS_H2 + (size_t)B_IMG * L2_BINS * 4;    // B*512  u32
static constexpr size_t WS_SCAL = WS_H3 + (size_t)B_IMG * L3_BINS * 4;    // 16 arrays of B u32
static constexpr size_t WS_IDX  = ((WS_SCAL + 16 * B_IMG * 4 + 255) / 256) * 256; // B*P int
static constexpr size_t WS_SCORE = WS_IDX + (size_t)B_IMG * P_PRE * 4;    // B*P f32
static constexpr size_t WS_BOX  = ((WS_SCORE + (size_t)B_IMG * P_PRE * 4 + 255) / 256) * 256; // B*P f32x4

// scalar sub-array indices within WS_SCAL (each B_IMG u32)
#define S_T1   0
#define S_A1   1
#define S_T2   2
#define S_A2   3
#define S_T3   4
#define S_AALL 5
#define S_KEY  6
#define S_CHI  7
#define S_CTIE 8

__device__ __forceinline__ unsigned score_key(float s) {
    unsigned u = __float_as_uint(s);
    return (u & 0x80000000u) ? ~u : (u | 0x80000000u);   // order-preserving float->u32
}
__device__ __forceinline__ float clamp01(float v) { return fminf(fmaxf(v, 0.0f), 1.0f); }

// ---------------- CDNA5 async global->LDS (ASYNCcnt) ----------------
__device__ __forceinline__ void async_load_b128_to_lds(unsigned lds_byte_addr,
                                                       const void* sbase,
                                                       unsigned byte_off) {
#if defined(__gfx1250__)
    asm volatile("global_load_async_to_lds_b128 %0, %1, %2"
                 :
                 : "v"(lds_byte_addr), "v"(byte_off), "s"(sbase)
                 : "memory");
#endif
}
__device__ __forceinline__ void wait_asynccnt0() {
#if defined(__gfx1250__)
    asm volatile("s_wait_asynccnt 0" ::: "memory");
#endif
}

// ---------------- kernels ----------------
__global__ void k_zero(unsigned* p, int n) {
    for (int i = blockIdx.x * blockDim.x + threadIdx.x; i < n; i += gridDim.x * blockDim.x)
        p[i] = 0u;
}

// level-1 histogram, LDS-privatized: NBLK blocks per image, 8KB LDS hist each,
// flushed once -> global atomic count drops from 5.3M to <= NBLK*B*2048.
#define H1_NBLK 36
__global__ __launch_bounds__(256) void k_hist1(const float* __restrict__ labels,
                                               unsigned* __restrict__ h1) {
    __shared__ unsigned lh[L1_BINS];
    for (int i = threadIdx.x; i < L1_BINS; i += 256) lh[i] = 0u;
    __syncthreads();
    const int b = blockIdx.x / H1_NBLK;
    const int chunk = blockIdx.x % H1_NBLK;
    const float* lab = labels + (size_t)b * N_ANCH;
    for (int i = chunk * 256 + threadIdx.x; i < N_ANCH; i += H1_NBLK * 256)
        atomicAdd(&lh[score_key(lab[i]) >> 21], 1u);
    __syncthreads();
    unsigned* hb = h1 + (size_t)b * L1_BINS;
    for (int i = threadIdx.x; i < L1_BINS; i += 256) {
        unsigned v = lh[i];
        if (v) atomicAdd(&hb[i], v);
    }
}

// level 2/3 histograms: only ~hist1[T1] (~tens of K) elements pass the filter,
// so plain global atomics are cheap here.
__global__ void k_hist23(const float* __restrict__ labels,
                         unsigned* __restrict__ h2, unsigned* __restrict__ h3,
                         const unsigned* __restrict__ scal, int level) {
    size_t gid = (size_t)blockIdx.x * blockDim.x + threadIdx.x;
    size_t total = (size_t)B_IMG * N_ANCH;
    if (gid >= total) return;
    int b = (int)(gid / N_ANCH);
    unsigned u = score_key(labels[gid]);
    if ((u >> 21) != scal[S_T1 * B_IMG + b]) return;
    unsigned b2 = (u >> 9) & 0xFFFu;
    if (level == 2) {
        atomicAdd(&h2[b * L2_BINS + b2], 1u);
    } else {
        if (b2 != scal[S_T2 * B_IMG + b]) return;
        atomicAdd(&h3[b * L3_BINS + (u & 0x1FFu)], 1u);
    }
}

// scan histograms from the top to locate the bin holding the P_PRE-th element
__global__ void k_scan(const unsigned* __restrict__ h1, const unsigned* __restrict__ h2,
                       const unsigned* __restrict__ h3, unsigned* __restrict__ scal,
                       int level) {
    int b = threadIdx.x;
    if (b >= B_IMG) return;
    if (level == 1) {
        unsigned cum = 0; int t = 0;
        for (int i = L1_BINS - 1; i >= 0; --i) {
            unsigned h = h1[b * L1_BINS + i];
            if (cum + h >= (unsigned)P_PRE) { t = i; break; }
            cum += h;
        }
        scal[S_T1 * B_IMG + b] = (unsigned)t;
        scal[S_A1 * B_IMG + b] = cum;
    } else if (level == 2) {
        unsigned target = (unsigned)P_PRE - scal[S_A1 * B_IMG + b];
        unsigned cum = 0; int t = 0;
        for (int i = L2_BINS - 1; i >= 0; --i) {
            unsigned h = h2[b * L2_BINS + i];
            if (cum + h >= target) { t = i; break; }
            cum += h;
        }
        scal[S_T2 * B_IMG + b] = (unsigned)t;
        scal[S_A2 * B_IMG + b] = cum;
    } else {
        unsigned a1 = scal[S_A1 * B_IMG + b], a2 = scal[S_A2 * B_IMG + b];
        unsigned target = (unsigned)P_PRE - a1 - a2;
        unsigned cum = 0; int t = 0;
        for (int i = L3_BINS - 1; i >= 0; --i) {
            unsigned h = h3[b * L3_BINS + i];
            if (cum + h >= target) { t = i; break; }
            cum += h;
        }
        scal[S_T3 * B_IMG + b] = (unsigned)t;
        scal[S_AALL * B_IMG + b] = a1 + a2 + cum;
        scal[S_KEY * B_IMG + b] =
            (scal[S_T1 * B_IMG + b] << 21) | (scal[S_T2 * B_IMG + b] << 9) | (unsigned)t;
    }
}

// compact top-P_PRE indices/scores per image using the exact 32-bit threshold key
__global__ void k_compact(const float* __restrict__ labels, unsigned* __restrict__ scal,
                          int* __restrict__ cidx, float* __restrict__ cscore) {
    size_t gid = (size_t)blockIdx.x * blockDim.x + threadIdx.x;
    size_t total = (size_t)B_IMG * N_ANCH;
    if (gid >= total) return;
    int b = (int)(gid / N_ANCH);
    int i = (int)(gid - (size_t)b * N_ANCH);
    float s = labels[gid];
    unsigned u = score_key(s);
    unsigned thr = scal[S_KEY * B_IMG + b];
    int slot = -1;
    if (u > thr) {
        slot = (int)atomicAdd(&scal[S_CHI * B_IMG + b], 1u);
    } else if (u == thr) {
        unsigned r = atomicAdd(&scal[S_CTIE * B_IMG + b], 1u);
        unsigned base = scal[S_AALL * B_IMG + b];
        if (base + r < (unsigned)P_PRE) slot = (int)(base + r);
    }
    if (slot >= 0) {
        cidx[(size_t)b * P_PRE + slot] = i;
        cscore[(size_t)b * P_PRE + slot] = s;
    }
}

// decode only the selected candidates: anchor from index, deltas*variances, exp, clip
__global__ void k_decode(const float* __restrict__ deltas, const float* __restrict__ ba,
                         const int* __restrict__ cidx, float4* __restrict__ cbox) {
    int gid = blockIdx.x * blockDim.x + threadIdx.x;
    if (gid >= B_IMG * P_PRE) return;
    int b = gid / P_PRE;
    int idx = cidx[gid];
    int a = idx % A_ANCH;
    int cell = idx / A_ANCH;
    int x = cell % FW, y = cell / FW;
    float cy = ((float)y + 0.5f) * (1.0f / FH);
    float cx = ((float)x + 0.5f) * (1.0f / FW);
    float4 bav = ((const float4*)ba)[a];
    float ay1 = clamp01(cy + bav.x), ax1 = clamp01(cx + bav.y);
    float ay2 = clamp01(cy + bav.z), ax2 = clamp01(cx + bav.w);
    float4 d4 = ((const float4*)deltas)[(size_t)b * N_ANCH + idx];
    float d0 = d4.x * 0.1f, d1 = d4.y * 0.1f, d2 = d4.z * 0.2f, d3 = d4.w * 0.2f;
    float ah = ay2 - ay1, aw = ax2 - ax1;
    float acy = ay1 + 0.5f * ah, acx = ax1 + 0.5f * aw;
    float h = expf(d2) * ah, w = expf(d3) * aw;
    float ncy = d0 * ah + acy, ncx = d1 * aw + acx;
    cbox[gid] = make_float4(clamp01(ncy - 0.5f * h), clamp01(ncx - 0.5f * w),
                            clamp01(ncy + 0.5f * h), clamp01(ncx + 0.5f * w));
}

// greedy argmax NMS: one 1024-thread workgroup per image, candidates in registers,
// box carried through the reduction -> only 2 barriers per round.
#define NMS_T 1024
#define CPT 6  // ceil(6000/1024)
__global__ __launch_bounds__(NMS_T) void k_nms(const float4* __restrict__ cbox,
                                               const float* __restrict__ cscore,
                                               float* __restrict__ out) {
    const int b = blockIdx.x;
    const int t = threadIdx.x;
    __shared__ float4 stage[NMS_T];
    __shared__ float red_v[NMS_T / 32];
    __shared__ int red_i[NMS_T / 32];
    __shared__ float4 red_b[NMS_T / 32];
    __shared__ float4 bc_box;
    __shared__ float bc_s;
    __shared__ int bc_i;

    const float4* boxes_g = cbox + (size_t)b * P_PRE;
    const float* sc_g = cscore + (size_t)b * P_PRE;
    float* outb = out + (size_t)b * Q_POST * 4;

    float4 rbox[CPT];
    float rsc[CPT];
    float rar[CPT];

    unsigned lds_stage = (unsigned)(uintptr_t)(&stage[0]);

    // Stage candidate boxes global->LDS with CDNA5 async copies, then pull to registers.
    // Each lane writes and reads only its own LDS slot, so s_wait_asynccnt (wave-local)
    // ordering is sufficient -- no workgroup barrier needed.
    for (int k = 0; k < CPT; ++k) {
        int c = k * NMS_T + t;
        if (c < P_PRE) {
            async_load_b128_to_lds(lds_stage + (unsigned)(t * 16), boxes_g, (unsigned)(c * 16));
            wait_asynccnt0();
            float4 v = stage[t];
            rbox[k] = v;
            rsc[k] = sc_g[c];
            rar[k] = fmaxf(v.z - v.x, 0.0f) * fmaxf(v.w - v.y, 0.0f);
        } else {
            rbox[k] = make_float4(0.f, 0.f, 0.f, 0.f);
            rsc[k] = NEG_V;
            rar[k] = 0.0f;
        }
    }

    const int wid = t >> 5, lane = t & 31;
    for (int r = 0; r < Q_POST; ++r) {
        // local argmax over this lane's 6 candidates
        float bv = rsc[0];
        int kb = 0;
#pragma unroll
        for (int k = 1; k < CPT; ++k)
            if (rsc[k] > bv) { bv = rsc[k]; kb = k; }
        int bi = kb * NMS_T + t;
        float4 bb = rbox[kb];
        // wave32 shuffle reduction carrying (score, index, box); tie -> lower index
#pragma unroll
        for (int off = 16; off > 0; off >>= 1) {
            float ov = __shfl_xor(bv, off, 32);
            int oi = __shfl_xor(bi, off, 32);
            float o0 = __shfl_xor(bb.x, off, 32), o1 = __shfl_xor(bb.y, off, 32);
            float o2 = __shfl_xor(bb.z, off, 32), o3 = __shfl_xor(bb.w, off, 32);
            if (ov > bv || (ov == bv && oi < bi)) {
                bv = ov; bi = oi; bb.x = o0; bb.y = o1; bb.z = o2; bb.w = o3;
            }
        }
        if (lane == 0) { red_v[wid] = bv; red_i[wid] = bi; red_b[wid] = bb; }
        __syncthreads();
        if (wid == 0) {
            bv = red_v[lane];
            bi = red_i[lane];
            bb = red_b[lane];
#pragma unroll
            for (int off = 16; off > 0; off >>= 1) {
                float ov = __shfl_xor(bv, off, 32);
                int oi = __shfl_xor(bi, off, 32);
                float o0 = __shfl_xor(bb.x, off, 32), o1 = __shfl_xor(bb.y, off, 32);
                float o2 = __shfl_xor(bb.z, off, 32), o3 = __shfl_xor(bb.w, off, 32);
                if (ov > bv || (ov == bv && oi < bi)) {
                    bv = ov; bi = oi; bb.x = o0; bb.y = o1; bb.z = o2; bb.w = o3;
                }
            }
            if (lane == 0) { bc_s = bv; bc_i = bi; bc_box = bb; }
        }
        __syncthreads();

        float ms = bc_s;
        if (ms <= NEG_V * 0.5f) {  // everything left suppressed -> zero-pad remainder
            for (int j = r * 4 + t; j < Q_POST * 4; j += NMS_T) outb[j] = 0.0f;
            break;
        }
        int mi = bc_i;
        float4 mb = bc_box;
        if (t == (mi & (NMS_T - 1))) rsc[mi >> 10] = NEG_V;  // owner clears winner

        float barea = fmaxf(mb.z - mb.x, 0.0f) * fmaxf(mb.w - mb.y, 0.0f);
#pragma unroll
        for (int k = 0; k < CPT; ++k) {
            float4 v = rbox[k];
            float iy1 = fmaxf(mb.x, v.x), ix1 = fmaxf(mb.y, v.y);
            float iy2 = fminf(mb.z, v.z), ix2 = fminf(mb.w, v.w);
            float inter = fmaxf(iy2 - iy1, 0.0f) * fmaxf(ix2 - ix1, 0.0f);
            float iou = inter / (barea + rar[k] - inter + 1e-8f);
            if (iou > IOU_THR) rsc[k] = NEG_V;
        }
        if (t == 0) {
            outb[r * 4 + 0] = mb.x;
            outb[r * 4 + 1] = mb.y;
            outb[r * 4 + 2] = mb.z;
            outb[r * 4 + 3] = mb.w;
        }
        // red[]/bc_* rewrites next round are fenced by this round's two barriers
    }
}

// ---------------- launch ----------------
extern "C" void kernel_launch(void* const* d_in, const int* in_sizes, int n_in,
                              void* d_out, int out_size, void* d_ws, size_t ws_size,
                              hipStream_t stream) {
    const float* deltas = (const float*)d_in[0];   // (B, FH, FW, A*4)
    const float* labels = (const float*)d_in[1];   // (B, FH, FW, A)
    const float* banch  = (const float*)d_in[2];   // (9, 4)
    float* out = (float*)d_out;                    // (B, 1000, 4)
    char* ws = (char*)d_ws;

    unsigned* h1   = (unsigned*)(ws + WS_H1);
    unsigned* h2   = (unsigned*)(ws + WS_H2);
    unsigned* h3   = (unsigned*)(ws + WS_H3);
    unsigned* scal = (unsigned*)(ws + WS_SCAL);
    int*      cidx   = (int*)(ws + WS_IDX);
    float*    cscore = (float*)(ws + WS_SCORE);
    float4*   cbox   = (float4*)(ws + WS_BOX);

    const int zero_n = (int)(WS_IDX / 4);  // hists + scalar counters
    k_zero<<<256, 256, 0, stream>>>((unsigned*)ws, zero_n);

    const size_t total = (size_t)B_IMG * N_ANCH;
    const int hblocks = (int)((total + 255) / 256);

    k_hist1<<<B_IMG * H1_NBLK, 256, 0, stream>>>(labels, h1);
    k_scan<<<1, B_IMG, 0, stream>>>(h1, h2, h3, scal, 1);
    k_hist23<<<hblocks, 256, 0, stream>>>(labels, h2, h3, scal, 2);
    k_scan<<<1, B_IMG, 0, stream>>>(h1, h2, h3, scal, 2);
    k_hist23<<<hblocks, 256, 0, stream>>>(labels, h2, h3, scal, 3);
    k_scan<<<1, B_IMG, 0, stream>>>(h1, h2, h3, scal, 3);
    k_compact<<<hblocks, 256, 0, stream>>>(labels, scal, cidx, cscore);

    k_decode<<<(B_IMG * P_PRE + 255) / 256, 256, 0, stream>>>(deltas, banch, cidx, cbox);
    k_nms<<<B_IMG, NMS_T, 0, stream>>>(cbox, cscore, out);
}